// GenerativeModel_86277303042785
// MI455X (gfx1250) — compile-verified
//
#include <hip/hip_runtime.h>
#include <hip/hip_bf16.h>

typedef __attribute__((ext_vector_type(2))) float v2f;
typedef __attribute__((ext_vector_type(8))) float v8f;

#define KK 10
#define DD 8
#define NN 32
#define TT 200
#define ZROWS 208          // TT padded up to 13*16
#define ZSTR 9             // row stride in LDS (pad to avoid bank conflicts)
#define NTILES 13          // ceil(200/16)
#define LOG2PI 1.8378770664093453f

__global__ __launch_bounds__(256) void slds_logjoint_kernel(
    const int* __restrict__ s,          // (B,T)
    const float* __restrict__ z,        // (B,T,D)
    const float* __restrict__ obs,      // (B,T,N)
    const float* __restrict__ init_logits,   // (K)
    const float* __restrict__ init_locs,     // (K,D)
    const float* __restrict__ init_ls,       // (K,D)
    const float* __restrict__ trans_logits,  // (K,K)
    const float* __restrict__ dynA,          // (K,D,D)
    const float* __restrict__ dynb,          // (K,D)
    const float* __restrict__ dynls,         // (K,D)
    const float* __restrict__ emC,           // (N,D)
    const float* __restrict__ emoff,         // (N)
    const float* __restrict__ emls,          // (N)
    float* __restrict__ out)                 // (B)
{
    __shared__ float z_s[ZROWS * ZSTR];
    __shared__ float dynA_s[KK * DD * DD];
    __shared__ float dynb_s[KK * DD];
    __shared__ float dynls_s[KK * DD];
    __shared__ float dyninv_s[KK * DD];
    __shared__ float trans_s[KK * KK];
    __shared__ float initlp_s[KK];
    __shared__ float initloc_s[KK * DD];
    __shared__ float initls_s[KK * DD];
    __shared__ float emoff_s[NN];
    __shared__ float emls_s[NN];
    __shared__ float eminv_s[NN];
    __shared__ float red_s[8];

    const int b    = blockIdx.x;
    const int tid  = threadIdx.x;
    const int lane = tid & 31;
    const int wave = tid >> 5;

    // ---------------- Phase 0: stage constants + z[b] into LDS ----------------
    for (int i = tid; i < KK * DD * DD; i += 256) dynA_s[i] = dynA[i];
    for (int i = tid; i < KK * DD; i += 256) {
        dynb_s[i] = dynb[i];
        float ls  = dynls[i];
        dynls_s[i]  = ls;
        dyninv_s[i] = __expf(-ls);
        initloc_s[i] = init_locs[i];
        initls_s[i]  = init_ls[i];
    }
    for (int i = tid; i < NN; i += 256) {
        emoff_s[i] = emoff[i];
        float ls   = emls[i];
        emls_s[i]  = ls;
        eminv_s[i] = __expf(-ls);
    }
    const float* zb = z + (size_t)b * TT * DD;
    for (int i = tid; i < TT * DD; i += 256)
        z_s[(i / DD) * ZSTR + (i % DD)] = zb[i];
    if (tid < (ZROWS - TT) * DD)   // zero-pad rows 200..207
        z_s[(TT + tid / DD) * ZSTR + (tid % DD)] = 0.0f;

    if (tid < KK) {
        // row-wise log-softmax of transition logits
        float m = -1e30f;
        for (int j = 0; j < KK; ++j) m = fmaxf(m, trans_logits[tid * KK + j]);
        float ssum = 0.0f;
        for (int j = 0; j < KK; ++j) ssum += __expf(trans_logits[tid * KK + j] - m);
        float lse = m + __logf(ssum);
        for (int j = 0; j < KK; ++j) trans_s[tid * KK + j] = trans_logits[tid * KK + j] - lse;
    } else if (tid == KK) {
        float m = -1e30f;
        for (int j = 0; j < KK; ++j) m = fmaxf(m, init_logits[j]);
        float ssum = 0.0f;
        for (int j = 0; j < KK; ++j) ssum += __expf(init_logits[j] - m);
        float lse = m + __logf(ssum);
        for (int j = 0; j < KK; ++j) initlp_s[j] = init_logits[j] - lse;
    }
    __syncthreads();

    // ---------------- Phase 1: emission term via V_WMMA_F32_16X16X4_F32 -------
    // A tile: 16 timesteps x 8 latent dims (two chained K=4 WMMAs)
    // B tile: emission_matrix^T (8 x 16), two column tiles cover N=32
    // f32 16x4 A layout: M = lane%16 ; K = {0,1} lanes 0-15, {2,3} lanes 16-31
    const int nlo = lane & 15;
    const int kb  = (lane >= 16) ? 2 : 0;

    v2f b00, b01, b10, b11;
    b00[0] = emC[nlo * DD + kb];            b00[1] = emC[nlo * DD + kb + 1];
    b01[0] = emC[nlo * DD + kb + 4];        b01[1] = emC[nlo * DD + kb + 5];
    b10[0] = emC[(nlo + 16) * DD + kb];     b10[1] = emC[(nlo + 16) * DD + kb + 1];
    b11[0] = emC[(nlo + 16) * DD + kb + 4]; b11[1] = emC[(nlo + 16) * DD + kb + 5];

    float acc = 0.0f;
    const float* ob = obs + (size_t)b * TT * NN;
    const int rowoff = (lane >= 16) ? 8 : 0;

    for (int tile = wave; tile < NTILES; tile += 8) {   // wave-uniform: EXEC all-ones
        const int tb = tile * 16;
        const int ta = tb + nlo;                         // A-fragment row (timestep)
        v2f a0, a1;
        a0[0] = z_s[ta * ZSTR + kb];
        a0[1] = z_s[ta * ZSTR + kb + 1];
        a1[0] = z_s[ta * ZSTR + kb + 4];
        a1[1] = z_s[ta * ZSTR + kb + 5];

        v8f c0 = {};
        c0 = __builtin_amdgcn_wmma_f32_16x16x4_f32(false, a0, false, b00, (short)0, c0, false, false);
        c0 = __builtin_amdgcn_wmma_f32_16x16x4_f32(false, a1, false, b01, (short)0, c0, false, false);
        v8f c1 = {};
        c1 = __builtin_amdgcn_wmma_f32_16x16x4_f32(false, a0, false, b10, (short)0, c1, false, false);
        c1 = __builtin_amdgcn_wmma_f32_16x16x4_f32(false, a1, false, b11, (short)0, c1, false, false);

        #pragma unroll
        for (int r = 0; r < 8; ++r) {
            const int trow  = tb + rowoff + r;           // timestep of this accumulator row
            const bool valid = (trow < TT);
            const int orow  = valid ? trow : (TT - 1);
            // column tile 0: n = nlo
            {
                const int col = nlo;
                float loc = c0[r] + emoff_s[col];
                float e   = ob[(size_t)orow * NN + col];
                float zz  = (e - loc) * eminv_s[col];
                float contrib = -0.5f * (zz * zz + LOG2PI) - emls_s[col];
                acc += valid ? contrib : 0.0f;
            }
            // column tile 1: n = nlo + 16
            {
                const int col = nlo + 16;
                float loc = c1[r] + emoff_s[col];
                float e   = ob[(size_t)orow * NN + col];
                float zz  = (e - loc) * eminv_s[col];
                float contrib = -0.5f * (zz * zz + LOG2PI) - emls_s[col];
                acc += valid ? contrib : 0.0f;
            }
        }
    }

    // ---------------- Phase 2: transition + dynamics (VALU from LDS) ----------
    const int* sb = s + (size_t)b * TT;
    for (int t = 1 + tid; t < TT; t += 256) {            // t = 1..199, one per thread
        const int sp = sb[t - 1];
        const int sc = sb[t];
        acc += trans_s[sp * KK + sc];
        const float* A  = &dynA_s[sc * DD * DD];
        const float* zp = &z_s[(t - 1) * ZSTR];
        const float* zc = &z_s[t * ZSTR];
        #pragma unroll
        for (int d = 0; d < DD; ++d) {
            float loc = dynb_s[sc * DD + d];
            #pragma unroll
            for (int i = 0; i < DD; ++i) loc += A[d * DD + i] * zp[i];
            float zz = (zc[d] - loc) * dyninv_s[sc * DD + d];
            acc += -0.5f * (zz * zz + LOG2PI) - dynls_s[sc * DD + d];
        }
    }

    // ---------------- Phase 3: init term (single thread) ----------------------
    if (tid == 255) {
        const int s0 = sb[0];
        acc += initlp_s[s0];
        #pragma unroll
        for (int d = 0; d < DD; ++d) {
            float ls = initls_s[s0 * DD + d];
            float zz = (z_s[d] - initloc_s[s0 * DD + d]) * __expf(-ls);
            acc += -0.5f * (zz * zz + LOG2PI) - ls;
        }
    }

    // ---------------- Final: block reduction -> out[b] ------------------------
    #pragma unroll
    for (int m = 16; m >= 1; m >>= 1) acc += __shfl_xor(acc, m, 32);
    if (lane == 0) red_s[wave] = acc;
    __syncthreads();
    if (tid == 0) {
        float tot = 0.0f;
        #pragma unroll
        for (int w = 0; w < 8; ++w) tot += red_s[w];
        out[b] = tot;
    }
}

extern "C" void kernel_launch(void* const* d_in, const int* in_sizes, int n_in,
                              void* d_out, int out_size, void* d_ws, size_t ws_size,
                              hipStream_t stream) {
    (void)in_sizes; (void)n_in; (void)d_ws; (void)ws_size; (void)out_size;
    const int*   s           = (const int*)  d_in[0];
    const float* z           = (const float*)d_in[1];
    const float* obs         = (const float*)d_in[2];
    const float* init_logits = (const float*)d_in[3];
    const float* init_locs   = (const float*)d_in[4];
    const float* init_ls     = (const float*)d_in[5];
    const float* trans       = (const float*)d_in[6];
    const float* dynA        = (const float*)d_in[7];
    const float* dynb        = (const float*)d_in[8];
    const float* dynls       = (const float*)d_in[9];
    const float* emC         = (const float*)d_in[10];
    const float* emoff       = (const float*)d_in[11];
    const float* emls        = (const float*)d_in[12];
    float* out = (float*)d_out;

    const int B = 4096;
    slds_logjoint_kernel<<<B, 256, 0, stream>>>(
        s, z, obs, init_logits, init_locs, init_ls, trans,
        dynA, dynb, dynls, emC, emoff, emls, out);
}